// Model1_18657337934163
// MI455X (gfx1250) — compile-verified
//
#include <hip/hip_runtime.h>
#include <hip/hip_bf16.h>
#include <math.h>

typedef __attribute__((ext_vector_type(2))) float v2f;
typedef __attribute__((ext_vector_type(8))) float v8f;

// ---------------------------------------------------------------------------
// Kernel 1: fold the batch-invariant circuit into a 16x16 complex unitary U,
// stored pre-swizzled for the f32 WMMA A-operand layout, and precompute the
// measurement+linear coefficient tensor C[c,p,k] = sum_w W[c,4w+p]*sign_w(k).
// Runs once, 1 block; state kept in registers via full unrolling.
// ---------------------------------------------------------------------------

__device__ __forceinline__ void apply1q(float sr[16], float si[16], int stv,
                                        float m00r, float m00i, float m01r, float m01i,
                                        float m10r, float m10i, float m11r, float m11i) {
#pragma unroll
    for (int k = 0; k < 16; ++k) {
        if (k & stv) continue;
        int k2 = k | stv;
        float ar = sr[k], ai = si[k], br = sr[k2], bi = si[k2];
        sr[k]  = m00r * ar - m00i * ai + m01r * br - m01i * bi;
        si[k]  = m00r * ai + m00i * ar + m01r * bi + m01i * br;
        sr[k2] = m10r * ar - m10i * ai + m11r * br - m11i * bi;
        si[k2] = m10r * ai + m10i * ar + m11r * bi + m11i * br;
    }
}

__device__ __forceinline__ void applyc1q(float sr[16], float si[16], int cs, int ts,
                                         float m00r, float m00i, float m01r, float m01i,
                                         float m10r, float m10i, float m11r, float m11i) {
#pragma unroll
    for (int k = 0; k < 16; ++k) {
        if (!(k & cs) || (k & ts)) continue;
        int k2 = k | ts;
        float ar = sr[k], ai = si[k], br = sr[k2], bi = si[k2];
        sr[k]  = m00r * ar - m00i * ai + m01r * br - m01i * bi;
        si[k]  = m00r * ai + m00i * ar + m01r * bi + m01i * br;
        sr[k2] = m10r * ar - m10i * ai + m11r * br - m11i * bi;
        si[k2] = m10r * ai + m10i * ar + m11r * bi + m11i * br;
    }
}

__device__ __forceinline__ void u3mat(float th, float ph, float lam,
                                      float& m00r, float& m00i, float& m01r, float& m01i,
                                      float& m10r, float& m10i, float& m11r, float& m11i) {
    float ct = cosf(0.5f * th), st = sinf(0.5f * th);
    m00r = ct;                  m00i = 0.0f;
    m01r = -cosf(lam) * st;     m01i = -sinf(lam) * st;
    m10r = cosf(ph) * st;       m10i = sinf(ph) * st;
    m11r = cosf(ph + lam) * ct; m11i = sinf(ph + lam) * ct;
}

__global__ void prep_kernel(const float* __restrict__ u3p,   // (5,4,3)
                            const float* __restrict__ cu3p,  // (5,3,3)
                            const float* __restrict__ W,     // (4,16)
                            float* __restrict__ Are,         // 256 f32, A-layout
                            float* __restrict__ Aim,         // 256 f32, A-layout
                            float* __restrict__ Cc)          // 256 f32: [(p*4+c)*16+m]
{
    int tid = threadIdx.x;

    if (tid < 16) {
        // Evolve basis column |tid> through the 5 blocks of u3/cu3 gates.
        float sr[16], si[16];
#pragma unroll
        for (int k = 0; k < 16; ++k) { sr[k] = 0.0f; si[k] = 0.0f; }
        sr[tid] = 1.0f;

        for (int blk = 0; blk < 5; ++blk) {
#pragma unroll
            for (int w = 0; w < 4; ++w) {
                const float* pp = u3p + (blk * 4 + w) * 3;
                float a, b, c, d, e, f, g, h;
                u3mat(pp[0], pp[1], pp[2], a, b, c, d, e, f, g, h);
                apply1q(sr, si, 8 >> w, a, b, c, d, e, f, g, h);
            }
#pragma unroll
            for (int w = 0; w < 3; ++w) {
                const float* pp = cu3p + (blk * 3 + w) * 3;
                float a, b, c, d, e, f, g, h;
                u3mat(pp[0], pp[1], pp[2], a, b, c, d, e, f, g, h);
                applyc1q(sr, si, 8 >> w, 4 >> w, a, b, c, d, e, f, g, h);
            }
        }

        // Scatter column j=tid into WMMA f32 16x16x4 A-operand layout.
        // Chunk q = j/4; within chunk kk = j%4:
        //   kk=0 -> VGPR0(.x) lanes 0-15 ; kk=1 -> VGPR1(.y) lanes 0-15
        //   kk=2 -> VGPR0(.x) lanes 16-31; kk=3 -> VGPR1(.y) lanes 16-31
        int q = tid >> 2, kk = tid & 3;
#pragma unroll
        for (int m = 0; m < 16; ++m) {
            int lane = m + ((kk >> 1) << 4);
            int idx = (q * 32 + lane) * 2 + (kk & 1);
            Are[idx] = sr[m];
            Aim[idx] = si[m];
        }
    }

    // C[c,p,m] = sum_w W[c, 4w+p] * sign_w(m), sign_w(m) = +1 if bit(3-w)==0
    {
        int idx = tid;          // 256 threads -> 256 entries
        int m = idx & 15;
        int c = (idx >> 4) & 3;
        int p = idx >> 6;
        float s = 0.0f;
#pragma unroll
        for (int w = 0; w < 4; ++w) {
            float sgn = ((m >> (3 - w)) & 1) ? -1.0f : 1.0f;
            s += W[c * 16 + 4 * w + p] * sgn;
        }
        Cc[idx] = s;
    }
}

// ---------------------------------------------------------------------------
// Kernel 2: fused pool -> cos/sin factors -> (U_re,U_im) @ V via WMMA ->
// probs -> measurement/linear fold -> log_softmax. 128 threads = 4 waves,
// 16 images per block (each wave: 4 images x 4 patches = 16 WMMA columns).
// ---------------------------------------------------------------------------
__global__ void __launch_bounds__(128) qcircuit_kernel(
        const float* __restrict__ x,
        const float* __restrict__ Are,
        const float* __restrict__ Aim,
        const float* __restrict__ Cc,
        const float* __restrict__ bias,
        float* __restrict__ out,
        int bsz)
{
    __shared__ float xpL[256];   // 16 images x 16 pooled values (i*4+j)
    __shared__ float ccL[256];

    int tid = threadIdx.x;
    ccL[tid]       = Cc[tid];
    ccL[tid + 128] = Cc[tid + 128];

    // ---- Phase 1: 6x6 mean pooling (2 pooled values per thread) ----
#pragma unroll
    for (int rep = 0; rep < 2; ++rep) {
        int q = tid + rep * 128;
        int img = q >> 4;
        int i = (q >> 2) & 3;
        int j = q & 3;
        int b = blockIdx.x * 16 + img;
        float sum = 0.0f;
        if (b < bsz) {
            const float* xb = x + (size_t)b * 784;
#pragma unroll
            for (int u = 0; u < 6; ++u) {
                // element offset is even -> 8B aligned float2 loads
                const v2f* row = (const v2f*)(xb + (6 * i + u) * 28 + 6 * j);
                v2f t0 = row[0], t1 = row[1], t2 = row[2];
                sum += t0.x + t0.y + t1.x + t1.y + t2.x + t2.y;
            }
        }
        xpL[q] = sum * (1.0f / 36.0f);
    }
    __syncthreads();

    // ---- Phase 2: per-lane column setup ----
    int lane = tid & 31;
    int wv = tid >> 5;
    int cl = lane & 15;                  // WMMA column (N) this lane serves
    int imgl = (wv << 2) + (cl >> 2);    // local image 0..15
    int p = cl & 3;                      // patch 0..3
    int cb = (p >> 1) << 1;              // c in {0,2}
    int rb = (p & 1) << 1;               // r in {0,2}

    float a0 = xpL[imgl * 16 + cb * 4 + rb];
    float a1 = xpL[imgl * 16 + cb * 4 + rb + 1];
    float a2 = xpL[imgl * 16 + (cb + 1) * 4 + rb];
    float a3 = xpL[imgl * 16 + (cb + 1) * 4 + rb + 1];

    float c0 = cosf(0.5f * a0), s0 = sinf(0.5f * a0);
    float c1 = cosf(0.5f * a1), s1 = sinf(0.5f * a1);
    float c2 = cosf(0.5f * a2), s2 = sinf(0.5f * a2);
    float c3 = cosf(0.5f * a3), s3 = sinf(0.5f * a3);

    // ---- Phase 3: f = U @ v for 16 columns via 8 chained f32 WMMAs ----
    // B element k = 4q + 2*hi + t decomposes as bits (i0,i1,i2,i3) =
    // (q>>1, q&1, hi, t): q,t are compile-time, only hi is a runtime select.
    int hi = (lane >> 4) & 1;            // lanes 16-31 carry K%4 in {2,3}
    float g2 = hi ? s2 : c2;             // single runtime select (wire 2)
    const v2f* A2r = (const v2f*)Are;
    const v2f* A2i = (const v2f*)Aim;

    v8f dre = {0.f, 0.f, 0.f, 0.f, 0.f, 0.f, 0.f, 0.f};
    v8f dim_ = {0.f, 0.f, 0.f, 0.f, 0.f, 0.f, 0.f, 0.f};
#pragma unroll
    for (int q = 0; q < 4; ++q) {
        v2f ar = A2r[q * 32 + lane];
        v2f ai = A2i[q * 32 + lane];
        float fa = (q >> 1) ? s0 : c0;   // compile-time select after unroll
        float fb = (q & 1) ? s1 : c1;    // compile-time select after unroll
        float common = fa * fb * g2;
        v2f bv;
        bv.x = common * c3;              // t = 0 (wire 3 = cos)
        bv.y = common * s3;              // t = 1 (wire 3 = sin)
        dre  = __builtin_amdgcn_wmma_f32_16x16x4_f32(false, ar, false, bv, (short)0, dre,  false, false);
        dim_ = __builtin_amdgcn_wmma_f32_16x16x4_f32(false, ai, false, bv, (short)0, dim_, false, false);
    }

    // ---- Phase 4: probs -> folded measurement+linear coefficients ----
    int mbase = hi * 8;                  // D rows: lanes<16 -> M=0..7, else 8..15
    float acc0 = 0.f, acc1 = 0.f, acc2 = 0.f, acc3 = 0.f;
#pragma unroll
    for (int r = 0; r < 8; ++r) {
        float pr = dre[r] * dre[r] + dim_[r] * dim_[r];
        int m = mbase + r;
        acc0 += ccL[(p * 4 + 0) * 16 + m] * pr;
        acc1 += ccL[(p * 4 + 1) * 16 + m] * pr;
        acc2 += ccL[(p * 4 + 2) * 16 + m] * pr;
        acc3 += ccL[(p * 4 + 3) * 16 + m] * pr;
    }
    // join the two M-halves of each column
    acc0 += __shfl_xor(acc0, 16);
    acc1 += __shfl_xor(acc1, 16);
    acc2 += __shfl_xor(acc2, 16);
    acc3 += __shfl_xor(acc3, 16);
    // sum the 4 patches of each image (columns 4i..4i+3)
    acc0 += __shfl_xor(acc0, 1); acc0 += __shfl_xor(acc0, 2);
    acc1 += __shfl_xor(acc1, 1); acc1 += __shfl_xor(acc1, 2);
    acc2 += __shfl_xor(acc2, 1); acc2 += __shfl_xor(acc2, 2);
    acc3 += __shfl_xor(acc3, 1); acc3 += __shfl_xor(acc3, 2);

    // ---- Phase 5: bias + log_softmax + store ----
    float l0 = acc0 + bias[0];
    float l1 = acc1 + bias[1];
    float l2 = acc2 + bias[2];
    float l3 = acc3 + bias[3];
    float mx = fmaxf(fmaxf(l0, l1), fmaxf(l2, l3));
    float se = expf(l0 - mx) + expf(l1 - mx) + expf(l2 - mx) + expf(l3 - mx);
    float lse = mx + logf(se);

    if (lane < 16 && (cl & 3) == 0) {
        int b = blockIdx.x * 16 + imgl;
        if (b < bsz) {
            float4 o;
            o.x = l0 - lse; o.y = l1 - lse; o.z = l2 - lse; o.w = l3 - lse;
            ((float4*)out)[b] = o;
        }
    }
}

// ---------------------------------------------------------------------------
extern "C" void kernel_launch(void* const* d_in, const int* in_sizes, int n_in,
                              void* d_out, int out_size, void* d_ws, size_t ws_size,
                              hipStream_t stream) {
    const float* x    = (const float*)d_in[0];
    const float* u3p  = (const float*)d_in[1];
    const float* cu3p = (const float*)d_in[2];
    const float* W    = (const float*)d_in[3];
    const float* bias = (const float*)d_in[4];
    float* out = (float*)d_out;

    float* wsf = (float*)d_ws;
    float* Are = wsf;          // 256 f32
    float* Aim = wsf + 256;    // 256 f32
    float* Cc  = wsf + 512;    // 256 f32

    int bsz = in_sizes[0] / 784;

    prep_kernel<<<1, 256, 0, stream>>>(u3p, cu3p, W, Are, Aim, Cc);

    int grid = (bsz + 15) / 16;
    qcircuit_kernel<<<grid, 128, 0, stream>>>(x, Are, Aim, Cc, bias, out, bsz);
}